// GCN_Standard_4028679324279
// MI455X (gfx1250) — compile-verified
//
#include <hip/hip_runtime.h>
#include <hip/hip_bf16.h>
#include <type_traits>

// GCN 2-layer forward for gfx1250 (MI455X).
// Strategy: CSR-by-destination gather (no float atomics), fp32 WMMA GEMMs.
// ws layout (4-byte units):
//   [counts:N][cursor:N][rowptr:N][bsums:NB][esorted:E][wsrc:E]
//   [dinv:N][H1:N*128][AGG1/h:N*128][H2:N*64]

typedef __attribute__((ext_vector_type(2))) float v2f;
typedef __attribute__((ext_vector_type(8))) float v8f;

#define DIN 128

// ---------------- small utility kernels ----------------
__global__ __launch_bounds__(256) void k_zero_int(int* __restrict__ p, int n) {
  int i = blockIdx.x * 256 + threadIdx.x;
  if (i < n) p[i] = 0;
}

__global__ __launch_bounds__(256) void k_hist(const int* __restrict__ dst,
                                              int* __restrict__ cnt, int E) {
  int e = blockIdx.x * 256 + threadIdx.x;
  if (e < E) atomicAdd(&cnt[dst[e]], 1);
}

// dinv[i] = rsqrt(in_degree + 1)   (self-loop guarantees deg >= 1)
__global__ __launch_bounds__(256) void k_dinv(const int* __restrict__ cnt,
                                              float* __restrict__ dinv, int n) {
  int i = blockIdx.x * 256 + threadIdx.x;
  if (i < n) dinv[i] = __frsqrt_rn((float)cnt[i] + 1.0f);
}

// ---------------- exclusive scan (deterministic, two-level) ----------------
__global__ __launch_bounds__(256) void k_block_scan(const int* __restrict__ cnt,
                                                    int* __restrict__ rowptr,
                                                    int* __restrict__ bsums, int n) {
  __shared__ int sh[256];
  int i = blockIdx.x * 256 + threadIdx.x;
  int v = (i < n) ? cnt[i] : 0;
  sh[threadIdx.x] = v;
  __syncthreads();
#pragma unroll
  for (int off = 1; off < 256; off <<= 1) {
    int t = (threadIdx.x >= off) ? sh[threadIdx.x - off] : 0;
    __syncthreads();
    sh[threadIdx.x] += t;
    __syncthreads();
  }
  if (i < n) rowptr[i] = sh[threadIdx.x] - v;  // exclusive
  if (threadIdx.x == 255) bsums[blockIdx.x] = sh[255];
}

__global__ __launch_bounds__(256) void k_scan_sums(int* __restrict__ bsums, int nb) {
  __shared__ int sh[256];
  int carry = 0;
  for (int base = 0; base < nb; base += 256) {
    int i = base + threadIdx.x;
    int v = (i < nb) ? bsums[i] : 0;
    sh[threadIdx.x] = v;
    __syncthreads();
#pragma unroll
    for (int off = 1; off < 256; off <<= 1) {
      int t = (threadIdx.x >= off) ? sh[threadIdx.x - off] : 0;
      __syncthreads();
      sh[threadIdx.x] += t;
      __syncthreads();
    }
    if (i < nb) bsums[i] = carry + sh[threadIdx.x] - v;  // exclusive
    carry += sh[255];
    __syncthreads();
  }
}

__global__ __launch_bounds__(256) void k_add_offsets(int* __restrict__ rowptr,
                                                     const int* __restrict__ bsums, int n) {
  int i = blockIdx.x * 256 + threadIdx.x;
  if (i < n) rowptr[i] += bsums[blockIdx.x];
}

// bucket-fill: esorted[rowptr[d] + slot] = src, wsrc[...] = dinv[src]
__global__ __launch_bounds__(256) void k_fill(const int* __restrict__ src,
                                              const int* __restrict__ dst,
                                              const int* __restrict__ rowptr,
                                              int* __restrict__ cursor,
                                              const float* __restrict__ dinv,
                                              int* __restrict__ esorted,
                                              float* __restrict__ wsrc, int E) {
  int e = blockIdx.x * 256 + threadIdx.x;
  if (e >= E) return;
  int s = src[e], d = dst[e];
  int j = rowptr[d] + atomicAdd(&cursor[d], 1);
  esorted[j] = s;
  wsrc[j] = dinv[s];
}

// ---------------- fp32 WMMA GEMM: C[M x 16*NT] = A[M x K] @ B[K x 16*NT] ----------------
// One wave computes a 16-row strip across the full output width (ISA §7.12.2 layouts).
template <int NT>
__global__ __launch_bounds__(256) void k_gemm_wmma_f32(const float* __restrict__ A,
                                                       const float* __restrict__ B,
                                                       float* __restrict__ C,
                                                       int M, int K) {
  constexpr int NCOL = 16 * NT;
  const int lane = threadIdx.x & 31;
  const int wave = threadIdx.x >> 5;
  const int mbase = (blockIdx.x * 8 + wave) * 16;
  if (mbase >= M) return;              // wave-uniform: EXEC stays all-ones for WMMA
  const int hi  = lane >> 4;           // 0: lanes 0-15, 1: lanes 16-31
  const int l15 = lane & 15;

  v8f acc[NT] = {};
  const float* __restrict__ arow = A + (size_t)(mbase + l15) * K;

  for (int k = 0; k < K; k += 4) {
    const int ka = k + 2 * hi;                 // even -> 8B aligned
    v2f a = *(const v2f*)(arow + ka);          // global_load_b64
#pragma unroll
    for (int t = 0; t < NT; ++t) {
      v2f b;
      b.x = B[(size_t)ka * NCOL + t * 16 + l15];
      b.y = B[(size_t)(ka + 1) * NCOL + t * 16 + l15];
      acc[t] = __builtin_amdgcn_wmma_f32_16x16x4_f32(
          /*neg_a=*/false, a, /*neg_b=*/false, b,
          /*c_mod=*/(short)0, acc[t], /*reuse_a=*/false, /*reuse_b=*/false);
    }
  }

#pragma unroll
  for (int t = 0; t < NT; ++t) {
#pragma unroll
    for (int r = 0; r < 8; ++r) {
      C[(size_t)(mbase + 8 * hi + r) * NCOL + t * 16 + l15] = acc[t][r];
    }
  }
}

// ---------------- CSR gather aggregation (wave per destination node) ----------------
// out[d] = [relu]( bias + H[d]*dinv[d]^2 + sum_j H[src_j]*dinv[src_j]*dinv[d] )
// Lane l owns channels [VPL*l, VPL*l+VPL). Edge metadata loaded once per 32-edge
// chunk and broadcast with wave32 __shfl; row gathers are coalesced b128/b64 loads
// served by the 192MB L2.
template <int C, bool RELU>
__global__ __launch_bounds__(256) void k_gather(const float* __restrict__ H,
                                                const int* __restrict__ rowptr,
                                                const int* __restrict__ cnt,
                                                const int* __restrict__ esorted,
                                                const float* __restrict__ wsrc,
                                                const float* __restrict__ dinv,
                                                const float* __restrict__ bias,
                                                float* __restrict__ out, int n) {
  constexpr int VPL = C / 32;  // 4 for C=128, 2 for C=64
  using VT = typename std::conditional<VPL == 4, float4, float2>::type;

  int d = blockIdx.x * 8 + (threadIdx.x >> 5);
  if (d >= n) return;
  int lane = threadIdx.x & 31;

  float dn = dinv[d];
  float acc[VPL];
  {
    VT t = *(const VT*)(H + (size_t)d * C + lane * VPL);  // self-loop term
    const float* tp = reinterpret_cast<const float*>(&t);
#pragma unroll
    for (int v = 0; v < VPL; ++v) acc[v] = tp[v] * dn * dn;
  }

  int start = rowptr[d];
  int len = cnt[d];
  for (int j0 = 0; j0 < len; j0 += 32) {
    int m = min(32, len - j0);
    int s_l = 0;
    float w_l = 0.0f;
    if (lane < m) {
      s_l = esorted[start + j0 + lane];
      w_l = wsrc[start + j0 + lane];
    }
    for (int i = 0; i < m; ++i) {
      int s = __shfl(s_l, i, 32);
      float w = __shfl(w_l, i, 32) * dn;
      VT t = *(const VT*)(H + (size_t)s * C + lane * VPL);
      const float* tp = reinterpret_cast<const float*>(&t);
#pragma unroll
      for (int v = 0; v < VPL; ++v) acc[v] += tp[v] * w;
    }
  }

#pragma unroll
  for (int v = 0; v < VPL; ++v) {
    float r = acc[v] + bias[lane * VPL + v];
    if (RELU) r = fmaxf(r, 0.0f);
    out[(size_t)d * C + lane * VPL + v] = r;
  }
}

// ---------------- host side ----------------
extern "C" void kernel_launch(void* const* d_in, const int* in_sizes, int n_in,
                              void* d_out, int out_size, void* d_ws, size_t ws_size,
                              hipStream_t stream) {
  (void)n_in; (void)out_size; (void)ws_size;
  const float* x    = (const float*)d_in[0];
  const int*   eidx = (const int*)d_in[1];   // [2, E]: row 0 = src, row 1 = dst
  const float* W1   = (const float*)d_in[2];
  const float* b1   = (const float*)d_in[3];
  const float* W2   = (const float*)d_in[4];
  const float* b2   = (const float*)d_in[5];
  float* out = (float*)d_out;

  const int N = in_sizes[0] / DIN;
  const int E = in_sizes[1] / 2;
  const int* esrc = eidx;
  const int* edst = eidx + E;

  const int gN  = (N + 255) / 256;   // also the number of scan blocks NB
  const int NB  = gN;
  const int gE  = (E + 255) / 256;
  const int gNw = (N + 7) / 8;                     // wave-per-node
  const int gGemm = (((N + 15) / 16) + 7) / 8;     // 8 strips of 16 rows per block

  // workspace carve-up
  int* wsi = (int*)d_ws;
  int* counts  = wsi;               // N
  int* cursor  = counts + N;        // N
  int* rowptr  = cursor + N;        // N
  int* bsums   = rowptr + N;        // NB
  int* esorted = bsums + NB;        // E
  float* wsrc  = (float*)(esorted + E);        // E
  float* dinv  = wsrc + E;                     // N
  float* H1    = dinv + N;                     // N*128
  float* AGG1  = H1 + (size_t)N * 128;         // N*128 (becomes h after fused bias+relu)
  float* H2    = AGG1 + (size_t)N * 128;       // N*64

  // --- build CSR by destination + normalization ---
  k_zero_int<<<gN, 256, 0, stream>>>(counts, N);
  k_zero_int<<<gN, 256, 0, stream>>>(cursor, N);
  k_hist<<<gE, 256, 0, stream>>>(edst, counts, E);
  k_dinv<<<gN, 256, 0, stream>>>(counts, dinv, N);
  k_block_scan<<<NB, 256, 0, stream>>>(counts, rowptr, bsums, N);
  k_scan_sums<<<1, 256, 0, stream>>>(bsums, NB);
  k_add_offsets<<<NB, 256, 0, stream>>>(rowptr, bsums, N);
  k_fill<<<gE, 256, 0, stream>>>(esrc, edst, rowptr, cursor, dinv, esorted, wsrc, E);

  // --- layer 1: H1 = x @ W1 ; h = relu(gather(H1) + b1) ---
  k_gemm_wmma_f32<8><<<gGemm, 256, 0, stream>>>(x, W1, H1, N, 128);
  k_gather<128, true><<<gNw, 256, 0, stream>>>(H1, rowptr, counts, esorted, wsrc,
                                               dinv, b1, AGG1, N);

  // --- layer 2: H2 = h @ W2 ; out = gather(H2) + b2 ---
  k_gemm_wmma_f32<4><<<gGemm, 256, 0, stream>>>(AGG1, W2, H2, N, 128);
  k_gather<64, false><<<gNw, 256, 0, stream>>>(H2, rowptr, counts, esorted, wsrc,
                                               dinv, b2, out, N);
}